// FullyConnectedDynamics_42399917146463
// MI455X (gfx1250) — compile-verified
//
#include <hip/hip_runtime.h>

typedef __attribute__((ext_vector_type(2))) float v2f;
typedef __attribute__((ext_vector_type(4))) float v4f;
typedef __attribute__((ext_vector_type(8))) float v8f;

#define KK   8
#define LL   16
#define LHH  64
#define CC   7                         // K-1
#define DD   (KK*LL + KK*CC*LHH)       // 3712
#define ROWB (DD*4)                    // row stride in bytes (14848)
#define ROWS 32                        // rows per wave (2 M-tiles of 16)
#define MT   2                         // M-tiles
#define ZK_STRIDE 132                  // floats; 528B rows (16B aligned, bank-shifted)
#define H_STRIDE  68                   // floats; 272B rows

static __device__ __forceinline__ v8f wmma4(v2f a, v2f b, v8f c) {
  // D = A(16x4,f32) * B(4x16,f32) + C(16x16,f32)
  return __builtin_amdgcn_wmma_f32_16x16x4_f32(
      /*neg_a=*/false, a, /*neg_b=*/false, b,
      /*c_mod=*/(short)0, c, /*reuse_a=*/false, /*reuse_b=*/false);
}

__global__ __launch_bounds__(32)
void fcd_kernel(const float* __restrict__ z,    // [B][3712]
                const float* __restrict__ Wz,   // [8][16][16]
                const float* __restrict__ bz,   // [8][16]
                const float* __restrict__ WH,   // [8][7][64][64]
                const float* __restrict__ bH,   // [8][7][64]
                const float* __restrict__ Wrz,  // [8][7][32][16]
                const float* __restrict__ brz,  // [8][7][32]
                const float* __restrict__ WrH,  // [8][7][16][32]
                const float* __restrict__ brH,  // [8][7][16]
                float* __restrict__ out)        // [B][3712]
{
  __shared__ float zk_lds[ROWS * ZK_STRIDE];
  __shared__ float h_lds [ROWS * H_STRIDE];
  __shared__ float h1_lds[ROWS * H_STRIDE];
  __shared__ float r_lds [ROWS * 32];

  const int lane = threadIdx.x & 31;
  const int row0 = blockIdx.x * ROWS;
  const int k    = blockIdx.y;
  const int n    = lane & 15;   // N column (B/C frags) and M row (A frags)
  const int h    = lane >> 4;   // half-wave selector

  const char* zbase = (const char*)z   + (size_t)row0 * ROWB;
  char*       obase = (char*)out       + (size_t)row0 * ROWB;
  char*       zkl   = (char*)&zk_lds[0];
  char*       hl    = (char*)&h_lds[0];
  char*       h1l   = (char*)&h1_lds[0];

  // ---- stage zK tile: 32 rows x 128 floats; base + immediate row offsets ----
  {
    const char* src = zbase + lane * 16;            // col = lane (v4f units)
    char*       dst = zkl + lane * 16;
    #pragma unroll
    for (int i = 0; i < 32; ++i)                    // row = i
      *(v4f*)(dst + i * (ZK_STRIDE*4)) = *(const v4f*)(src + i * ROWB);
  }

  // ---- z1 base: zK[:,k,:] @ Wz[k]^T  (za accumulates here too) ----
  v8f z1acc[MT];
  #pragma unroll
  for (int mt = 0; mt < MT; ++mt)
    z1acc[mt] = (v8f){0.f,0.f,0.f,0.f,0.f,0.f,0.f,0.f};
  {
    const char* wzp = (const char*)Wz + k*1024 + n*64 + h*8;       // Wz[k][n][2h]
    const char* azp = zkl + n*(ZK_STRIDE*4) + k*64 + h*8;          // zK[m][k*16+2h]
    #pragma unroll
    for (int t = 0; t < 4; ++t) {
      v2f b = *(const v2f*)(wzp + t*16);
      #pragma unroll
      for (int mt = 0; mt < MT; ++mt) {
        v2f a = *(const v2f*)(azp + mt*(16*ZK_STRIDE*4) + t*16);
        z1acc[mt] = wmma4(a, b, z1acc[mt]);
      }
    }
  }

  float brH_sum = 0.f;

  for (int c = 0; c < CC; ++c) {
    const int ct = (c < k) ? c : c + 1;
    const int kc = k * CC + c;

    // ---- stage H tile: 32 rows x 64 floats; base + immediate offsets ----
    // per lane: row = h + 2i, col(v4f) = n
    const char* hsrc = zbase + (size_t)h * ROWB + 512 + (size_t)kc * 256 + n * 16;
    {
      char* dst = hl + h*(H_STRIDE*4) + n*16;
      #pragma unroll
      for (int i = 0; i < 16; ++i)
        *(v4f*)(dst + i * (2*H_STRIDE*4)) = *(const v4f*)(hsrc + i * (2*ROWB));
    }

    // prefetch next c's H tile (global_prefetch_b8)
    if (c + 1 < CC) {
      __builtin_prefetch(zbase + (size_t)lane * ROWB + 512 + (size_t)(kc+1) * 256, 0, 1);
      __builtin_prefetch(zbase + (size_t)lane * ROWB + 512 + (size_t)(kc+1) * 256 + 128, 0, 1);
    }

    // ---- GEMM2: r = zK[:,ct,:] @ Wrz[k,c]^T   (32x32 out) ----
    {
      v8f racc[MT][2];
      #pragma unroll
      for (int mt = 0; mt < MT; ++mt)
        #pragma unroll
        for (int q = 0; q < 2; ++q)
          racc[mt][q] = (v8f){0.f,0.f,0.f,0.f,0.f,0.f,0.f,0.f};
      const char* wrzp = (const char*)Wrz + kc*2048 + n*64 + h*8;  // Wrz[kc][n][2h]
      const char* azp  = zkl + n*(ZK_STRIDE*4) + ct*64 + h*8;      // zK[m][ct*16+2h]
      #pragma unroll
      for (int t = 0; t < 4; ++t) {
        v2f b0 = *(const v2f*)(wrzp + t*16);                       // o = n
        v2f b1 = *(const v2f*)(wrzp + 1024 + t*16);                // o = 16+n
        #pragma unroll
        for (int mt = 0; mt < MT; ++mt) {
          v2f a = *(const v2f*)(azp + mt*(16*ZK_STRIDE*4) + t*16);
          racc[mt][0] = wmma4(a, b0, racc[mt][0]);
          racc[mt][1] = wmma4(a, b1, racc[mt][1]);
        }
      }
      const float br0 = brz[kc * 32 + n];
      const float br1 = brz[kc * 32 + 16 + n];
      float* rp = &r_lds[(8*h) * 32 + n];                          // m = mt*16+rr+8h
      #pragma unroll
      for (int mt = 0; mt < MT; ++mt)
        #pragma unroll
        for (int rr = 0; rr < 8; ++rr) {
          rp[(mt*16 + rr) * 32]      = racc[mt][0][rr] + br0;
          rp[(mt*16 + rr) * 32 + 16] = racc[mt][1][rr] + br1;
        }
    }

    // ---- GEMM1: H1 = H @ WH[k,c]^T   (32x64 out, K=64) ----
    v8f acc[MT][4];
    #pragma unroll
    for (int mt = 0; mt < MT; ++mt)
      #pragma unroll
      for (int q = 0; q < 4; ++q)
        acc[mt][q] = (v8f){0.f,0.f,0.f,0.f,0.f,0.f,0.f,0.f};
    {
      const char* whp = (const char*)WH + (size_t)kc*16384 + n*256 + h*8; // WH[kc][n][2h]
      const char* ap  = hl + n*(H_STRIDE*4) + h*8;
      #pragma unroll
      for (int t = 0; t < 16; ++t) {
        v2f a0 = *(const v2f*)(ap + t*16);
        v2f a1 = *(const v2f*)(ap + 16*(H_STRIDE*4) + t*16);
        #pragma unroll
        for (int q = 0; q < 4; ++q) {
          v2f b = *(const v2f*)(whp + q*4096 + t*16);   // B[i][o]=WH[o][i]
          acc[0][q] = wmma4(a0, b, acc[0][q]);          // B frag reused across M-tiles
          acc[1][q] = wmma4(a1, b, acc[1][q]);
        }
      }
    }

    // ---- compose H1 = acc + bH (+ r on even cols) into LDS ----
    {
      const float* bhp = bH + (size_t)kc * 64;
      const bool even = ((n & 1) == 0);
      const float* rp  = &r_lds[(8*h) * 32 + (n >> 1)];
      char*        op  = h1l + (8*h)*(H_STRIDE*4) + n*4;
      #pragma unroll
      for (int q = 0; q < 4; ++q) {
        const float bias = bhp[q*16 + n];
        #pragma unroll
        for (int mt = 0; mt < MT; ++mt)
          #pragma unroll
          for (int rr = 0; rr < 8; ++rr) {
            float v = acc[mt][q][rr] + bias;
            if (even) v += rp[(mt*16 + rr) * 32 + q*8];
            *(float*)(op + (mt*16 + rr)*(H_STRIDE*4) + q*64) = v;
          }
      }
    }

    // ---- GEMM3: z1acc += H1[...,::2] @ WrH[k,c]^T   (32x16 out, K=32) ----
    {
      const char* wrhp = (const char*)WrH + kc*2048 + n*128 + h*8;   // WrH[kc][n][2h]
      const char* ahp  = h1l + n*(H_STRIDE*4) + h*16;                // even col 2*(2h)
      #pragma unroll
      for (int t = 0; t < 8; ++t) {
        v2f b = *(const v2f*)(wrhp + t*16);
        #pragma unroll
        for (int mt = 0; mt < MT; ++mt) {
          v2f a;
          a.x = *(const float*)(ahp + mt*(16*H_STRIDE*4) + t*32);       // col 2*kk
          a.y = *(const float*)(ahp + mt*(16*H_STRIDE*4) + t*32 + 8);   // col 2*(kk+1)
          z1acc[mt] = wmma4(a, b, z1acc[mt]);
        }
      }
    }

    brH_sum += brH[kc * 16 + n];

    // ---- write H1 tile to out (coalesced v4f from LDS, immediate offsets) ----
    {
      char*       dst = obase + (size_t)h * ROWB + 512 + (size_t)kc * 256 + n * 16;
      const char* srcl = h1l + h*(H_STRIDE*4) + n*16;
      #pragma unroll
      for (int i = 0; i < 16; ++i)
        *(v4f*)(dst + i * (2*ROWB)) = *(const v4f*)(srcl + i * (2*H_STRIDE*4));
    }
  }

  // ---- finalize & store z1 (adds bz + sum_c brH) ----
  {
    const float bias = bz[k * 16 + n] + brH_sum;
    char* zop = obase + (size_t)(8*h) * ROWB + k*64 + n*4;
    #pragma unroll
    for (int mt = 0; mt < MT; ++mt)
      #pragma unroll
      for (int rr = 0; rr < 8; ++rr)
        *(float*)(zop + (size_t)(mt*16 + rr) * ROWB) = z1acc[mt][rr] + bias;
  }
}

extern "C" void kernel_launch(void* const* d_in, const int* in_sizes, int n_in,
                              void* d_out, int out_size, void* d_ws, size_t ws_size,
                              hipStream_t stream) {
  const float* z   = (const float*)d_in[0];
  const float* Wz  = (const float*)d_in[1];
  const float* bz  = (const float*)d_in[2];
  const float* WH  = (const float*)d_in[3];
  const float* bH  = (const float*)d_in[4];
  const float* Wrz = (const float*)d_in[5];
  const float* brz = (const float*)d_in[6];
  const float* WrH = (const float*)d_in[7];
  const float* brH = (const float*)d_in[8];
  float* out = (float*)d_out;

  const int Brows = in_sizes[0] / DD;            // 16384
  dim3 grid((Brows + ROWS - 1) / ROWS, KK);      // (512, 8)
  fcd_kernel<<<grid, dim3(32), 0, stream>>>(z, Wz, bz, WH, bH, Wrz, brz, WrH, brH, out);
}